// PQ_87540023427438
// MI455X (gfx1250) — compile-verified
//
#include <hip/hip_runtime.h>

typedef __attribute__((ext_vector_type(2))) float v2f;
typedef __attribute__((ext_vector_type(8))) float v8f;

#define PQ_B   16
#define PQ_D   128
#define PQ_M   16
#define PQ_KS  256
#define PQ_DS  8
#define DT_STRIDE (PQ_M * PQ_KS)          // 4096 floats per b-slice
#define DT_BYTES  (PQ_B * PQ_M * PQ_KS * 4)  // 256 KB

// dt layout: dt[b*4096 + m*256 + k]  -> stage-2 gathers at fixed (b,m) index a
// random k => bank = k mod 64, near conflict-free across the wave.
extern "C" __global__ __launch_bounds__(512, 1)
void pq_adc_wmma_kernel(const float* __restrict__ q,     // (16,128)
                        const float* __restrict__ cw,    // (16,256,8)
                        const int*   __restrict__ codes, // (N,16) int32
                        float* __restrict__ out,         // (16,N)
                        int N)
{
    extern __shared__ float dt[];

    const int tid  = threadIdx.x;
    const int lane = tid & 31;
    const int wave = tid >> 5;          // 0..15
    const int l16  = lane & 15;
    const int hi   = (lane >> 4) & 1;   // 0: lanes 0-15 (K pair {0,1}), 1: lanes 16-31 ({2,3})

    // ---------------- Phase 1: build 16x16x256 distance table with WMMA ----------------
    // Job j = (m, kt): 16x16 tile of dtable, D(b,k) = A(16x12) x B(12x16) via 3 chained
    // 16x16x4 f32 WMMAs.  A row b = [q_bm(8), ||q_bm||^2, 1, 0, 0],
    // B col k = [-2*c_mk(8), 1, ||c_mk||^2, 0, 0]^T  =>  D = squared L2 distance tile.
    for (int j = wave; j < PQ_M * (PQ_KS / 16); j += 16) {   // uniform per wave, EXEC all-1s
        const int m  = j >> 4;
        const int kt = j & 15;

        const float* qp = q + l16 * PQ_D + m * PQ_DS;        // A row = b = lane&15
        float4 qa = *(const float4*)(qp);
        float4 qb = *(const float4*)(qp + 4);
        const float* cp = cw + (size_t)(m * PQ_KS + kt * 16 + l16) * PQ_DS; // B col = k
        float4 c0 = *(const float4*)(cp);
        float4 c1 = *(const float4*)(cp + 4);

        const float qq = qa.x*qa.x + qa.y*qa.y + qa.z*qa.z + qa.w*qa.w
                       + qb.x*qb.x + qb.y*qb.y + qb.z*qb.z + qb.w*qb.w;
        const float cc = c0.x*c0.x + c0.y*c0.y + c0.z*c0.z + c0.w*c0.w
                       + c1.x*c1.x + c1.y*c1.y + c1.z*c1.z + c1.w*c1.w;

        // 32-bit A 16x4 layout: lanes 0-15 hold K={0,1}, lanes 16-31 K={2,3} (per row b).
        v2f A0, B0, A1, B1, A2, B2;
        A0.x = hi ? qa.z : qa.x;            A0.y = hi ? qa.w : qa.y;
        B0.x = -2.0f * (hi ? c0.z : c0.x);  B0.y = -2.0f * (hi ? c0.w : c0.y);
        A1.x = hi ? qb.z : qb.x;            A1.y = hi ? qb.w : qb.y;
        B1.x = -2.0f * (hi ? c1.z : c1.x);  B1.y = -2.0f * (hi ? c1.w : c1.y);
        // K = 8..11: [qq, 1 | 0, 0]  x  [1, cc | 0, 0]
        A2.x = hi ? 0.0f : qq;              A2.y = hi ? 0.0f : 1.0f;
        B2.x = hi ? 0.0f : 1.0f;            B2.y = hi ? 0.0f : cc;

        v8f acc = {};
        acc = __builtin_amdgcn_wmma_f32_16x16x4_f32(false, A0, false, B0, (short)0, acc, false, false);
        acc = __builtin_amdgcn_wmma_f32_16x16x4_f32(false, A1, false, B1, (short)0, acc, false, false);
        acc = __builtin_amdgcn_wmma_f32_16x16x4_f32(false, A2, false, B2, (short)0, acc, false, false);

        // C/D layout: VGPR r, lanes 0-15 -> M=r, lanes 16-31 -> M=r+8; N = lane&15.
        const int kcol = kt * 16 + l16;
        #pragma unroll
        for (int r = 0; r < 8; ++r) {
            const int b = r + hi * 8;
            dt[b * DT_STRIDE + m * PQ_KS + kcol] = acc[r];
        }
    }
    __syncthreads();

    // ---------------- Phase 2: gather + accumulate + sqrt (LDS-gather bound) ----------------
    for (int n = blockIdx.x * blockDim.x + tid; n < N; n += gridDim.x * blockDim.x) {
        const int4* cp4 = (const int4*)(codes + (size_t)n * PQ_M);  // 64B coalesced per lane
        const int4 ka = cp4[0];
        const int4 kb = cp4[1];
        const int4 kc = cp4[2];
        const int4 kd = cp4[3];

        int offs[16];
        offs[ 0] =  0 * PQ_KS + ka.x;  offs[ 1] =  1 * PQ_KS + ka.y;
        offs[ 2] =  2 * PQ_KS + ka.z;  offs[ 3] =  3 * PQ_KS + ka.w;
        offs[ 4] =  4 * PQ_KS + kb.x;  offs[ 5] =  5 * PQ_KS + kb.y;
        offs[ 6] =  6 * PQ_KS + kb.z;  offs[ 7] =  7 * PQ_KS + kb.w;
        offs[ 8] =  8 * PQ_KS + kc.x;  offs[ 9] =  9 * PQ_KS + kc.y;
        offs[10] = 10 * PQ_KS + kc.z;  offs[11] = 11 * PQ_KS + kc.w;
        offs[12] = 12 * PQ_KS + kd.x;  offs[13] = 13 * PQ_KS + kd.y;
        offs[14] = 14 * PQ_KS + kd.z;  offs[15] = 15 * PQ_KS + kd.w;

        #pragma unroll
        for (int b = 0; b < PQ_B; ++b) {
            const float* base = dt + b * DT_STRIDE;
            float s = 0.0f;
            #pragma unroll
            for (int m = 0; m < PQ_M; ++m) s += base[offs[m]];
            out[(size_t)b * N + n] = __builtin_sqrtf(s);   // coalesced in n
        }
    }
}

extern "C" void kernel_launch(void* const* d_in, const int* in_sizes, int n_in,
                              void* d_out, int out_size, void* d_ws, size_t ws_size,
                              hipStream_t stream) {
    (void)n_in; (void)out_size; (void)d_ws; (void)ws_size;
    const float* q     = (const float*)d_in[0];   // (16,128) f32
    const float* cw    = (const float*)d_in[1];   // (16,256,8) f32
    const int*   codes = (const int*)d_in[2];     // (N,16) int32
    float*       out   = (float*)d_out;           // (16,N) f32

    const int N = in_sizes[2] / PQ_M;             // 250000
    const int threads = 512;
    int blocks = (N + threads - 1) / threads;     // ~489: one pass per WG
    if (blocks < 1) blocks = 1;

    pq_adc_wmma_kernel<<<blocks, threads, DT_BYTES, stream>>>(q, cw, codes, out, N);
}